// Encoder_71554155151417
// MI455X (gfx1250) — compile-verified
//
#include <hip/hip_runtime.h>
#include <math.h>

typedef float v2f __attribute__((ext_vector_type(2)));
typedef float v8f __attribute__((ext_vector_type(8)));
typedef unsigned int v4u __attribute__((ext_vector_type(4)));
typedef int v4i __attribute__((ext_vector_type(4)));
typedef int v8i __attribute__((ext_vector_type(8)));

#define WMMA_F32(a, b, c) \
    __builtin_amdgcn_wmma_f32_16x16x4_f32(false, (a), false, (b), (short)0, (c), false, false)

// Problem dims (fixed by the reference)
static constexpr int Bb = 8;       // batch
static constexpr int Ss = 1024;    // sequence
static constexpr int Dd = 300;     // hidden
static constexpr int DhT = 512;    // qkv total dim
static constexpr int Hh = 8;       // heads
static constexpr int dh = 64;      // head dim
static constexpr int Df = 1200;    // ffn dim
static constexpr int Mm = Bb * Ss; // 8192 rows

// ---------------------------------------------------------------------------
// LDS helpers
// ---------------------------------------------------------------------------
__device__ __forceinline__ unsigned lds_addr32(const void* p) {
    // low 32 bits of a generic pointer into LDS == wave-relative LDS byte offset
    return (unsigned)(unsigned long long)p;
}

// TDM row-strided load: 16 rows of `kelems` f32 each, global rows contiguous
// (stride = kelems), LDS rows at stride `lds_stride` floats.  Implemented with
// a single iterated 1-D descriptor: iterate_enable with per-iteration
// global_addr_increment = kelems and lds_addr_increment = lds_stride.
__device__ __forceinline__ void tdm_load_rows16(unsigned lds_off, const void* gsrc,
                                                unsigned kelems, unsigned lds_stride) {
    unsigned long long ga = (unsigned long long)gsrc;
    v4u g0;
    g0.x = 1u;                                   // count=1, user mode, no gather
    g0.y = lds_off;                              // lds_addr (bytes)
    g0.z = (unsigned)ga;                         // global_addr[31:0]
    g0.w = (unsigned)((ga >> 32) & 0x01ffffffu)  // global_addr[56:32]
           | (2u << 30);                         // type = 2 ("image")
    v8i g1;
    g1[0] = (int)((2u << 16)                     // data_size = 4B
                  | (1u << 19));                 // iterate_enable
    g1[1] = (int)((kelems & 0xffffu) << 16);     // tensor_dim0[15:0]
    g1[2] = (int)(((kelems >> 16) & 0xffffu)     // tensor_dim0[31:16]
                  | (1u << 16));                 // tensor_dim1 = 1
    g1[3] = (int)((kelems & 0xffffu) << 16);     // tile_dim0 = kelems (one row)
    g1[4] = 0;                                   // tile_dim1/2 unused
    g1[5] = (int)kelems;                         // tensor_dim0_stride
    g1[6] = 0;
    g1[7] = 0;
    v4i g2;
    g2.x = 1;                                    // tensor_dim2 (don't care)
    g2.y = (int)lds_stride;                      // lds_addr_increment (elems)
    g2.z = (int)kelems;                          // global_addr_increment (elems)
    g2.w = (int)(15u << 16);                     // iterate_count = 15 -> 16 rows
    v4i gz = {};
#if defined(__clang_major__) && (__clang_major__ >= 23)
    v8i gz8 = {};
    __builtin_amdgcn_tensor_load_to_lds(g0, g1, g2, gz, gz8, 0);
#else
    __builtin_amdgcn_tensor_load_to_lds(g0, g1, g2, gz, 0);
#endif
}

// ---------------------------------------------------------------------------
// Shared WMMA mainloop: A[16 rows x K] in LDS at row stride LDA (gcd(LDA,64)=4
// -> conflict-free), B column from global with row stride LDB.
// ---------------------------------------------------------------------------
template <int K, int LDA, int LDB>
__device__ __forceinline__ v8f wmma_ldsA_loop(const float* __restrict__ smA,
                                              const float* __restrict__ bcol,
                                              int r, int hf) {
    const float* rowA = smA + r * LDA + 2 * hf;  // loop-invariant base
    v8f acc = {};
#pragma unroll 4
    for (int k0 = 0; k0 < K; k0 += 4) {
        v2f a = *(const v2f*)(rowA + k0);        // ds_load_b64, imm offset
        const int ka = k0 + 2 * hf;
        v2f b;
        b.x = bcol[(size_t)ka * LDB];
        b.y = bcol[(size_t)(ka + 1) * LDB];
        acc = WMMA_F32(a, b, acc);
    }
    return acc;
}

// ---------------------------------------------------------------------------
// Kernel 1: QKV projection.  C[8192,512] = X[8192,300] * W^T, W=[512,300].
// 4 waves per block; A strip (16x300) staged by TDM into LDS (LDA=300,
// 300 % 64 = 44, gcd 4 -> conflict-free).  Head-split on write.
// ---------------------------------------------------------------------------
static constexpr int LDA_300 = 300;
static constexpr int LDA_512 = 516;
static constexpr int LDA_1200 = 1204;

__global__ void qkv_kernel(const float* __restrict__ x,
                           const float* __restrict__ WQ,
                           const float* __restrict__ WK,
                           const float* __restrict__ WV,
                           float* __restrict__ qb,
                           float* __restrict__ kb,
                           float* __restrict__ vb) {
    extern __shared__ float smA[];
    const int tid = threadIdx.x;
    const int lane = tid & 31, w = tid >> 5;
    const int r = lane & 15, hf = lane >> 4;
    const int m0 = blockIdx.x * 16;
    const int n0 = blockIdx.y * 64 + w * 16;
    const float* W = (blockIdx.z == 0) ? WQ : ((blockIdx.z == 1) ? WK : WV);
    float* outb = (blockIdx.z == 0) ? qb : ((blockIdx.z == 1) ? kb : vb);

    if (tid < 32) {
        tdm_load_rows16(lds_addr32(smA), x + (size_t)m0 * Dd, Dd, LDA_300);
        __builtin_amdgcn_s_wait_tensorcnt(0);
    }
    __syncthreads();

    const int c = n0 + r;
    v8f acc = wmma_ldsA_loop<Dd, LDA_300, 1>(smA, W + (size_t)c * Dd, r, hf);

    const int hsel = c & 7;
    const int dsel = c >> 3;
    for (int i = 0; i < 8; ++i) {
        int m = m0 + i + 8 * hf;
        int bI = m >> 10;
        int sI = m & 1023;
        outb[(((size_t)hsel * Bb + bI) * Ss + sI) * dh + dsel] = acc[i];
    }
}

// ---------------------------------------------------------------------------
// Kernel 2: attention for one (head, batch, 16-row strip).
// 128 threads = 4 waves.  Q strip + double-buffered per-wave K tiles staged
// via async global->LDS (row stride 68: 16B-aligned, banks r*4 -> conflict-
// free).  e strip [16 x 1024] lives in LDS.
// ---------------------------------------------------------------------------
#define ESTRIDE 1025
#define QKSTRIDE 68   // floats; 272B rows: 16B aligned, stride % 64 == 4

__global__ void attn_kernel(const float* __restrict__ qb,
                            const float* __restrict__ kb,
                            const float* __restrict__ vb,
                            const float* __restrict__ gamma_a,
                            const float* __restrict__ beta_a,
                            float* __restrict__ ab) {
    extern __shared__ float sm[];
    float* e     = sm;                     // 16 * 1025
    float* red   = sm + 16 * ESTRIDE;      // 256
    float* rowMu = red + 256;              // 16
    float* rowRs = rowMu + 16;             // 16
    float* rowMx = rowRs + 16;             // 16
    float* rowSm = rowMx + 16;             // 16
    float* smQ   = rowSm + 16;             // 16 * QKSTRIDE
    float* smK   = smQ + 16 * QKSTRIDE;    // 4 waves * 2 bufs * 16 * QKSTRIDE

    const int tid = threadIdx.x;
    const int lane = tid & 31;
    const int w = tid >> 5;
    const int r = lane & 15;
    const int hf = lane >> 4;
    const int s0 = blockIdx.x * 16;
    const int bI = blockIdx.y;
    const int h = blockIdx.z;
    const float* qh = qb + ((size_t)(h * Bb + bI) * Ss) * dh;
    const float* kh = kb + ((size_t)(h * Bb + bI) * Ss) * dh;
    const float* vh = vb + ((size_t)(h * Bb + bI) * Ss) * dh;
    const float scale = 0.057735026918962584f;  // 1/sqrt(300)

    // Stage Q strip (16 x 64 f32 = 4KB contiguous).  Thread t moves 32B at
    // mem offset t*32; LDS dest = mem + 16*(t>>3) (row pad), so one base
    // register serves both addresses and offset: immediates advance both.
    {
        const unsigned qlds = lds_addr32(smQ);
        const unsigned long long qg = (unsigned long long)(qh + (size_t)s0 * dh);
        unsigned vo = (unsigned)(tid * 32);
        unsigned dd = qlds + vo + (unsigned)((tid >> 3) * 16);
        asm volatile(
            "global_load_async_to_lds_b128 %0, %1, %2 offset:0\n\t"
            "global_load_async_to_lds_b128 %0, %1, %2 offset:16"
            :: "v"(dd), "v"(vo), "s"(qg)
            : "memory");
        asm volatile("s_wait_asynccnt 0x0" ::: "memory");
    }
    __syncthreads();

    // Per-wave async K-tile fetch: lane moves 128B at mem offset lane*128;
    // LDS dest = mem + 16*(lane>>1)  (row = lane>>1, 272B rows).
    float* smKw = smK + w * (2 * 16 * QKSTRIDE);
    auto issueK = [&](int t0, float* dstbuf) {
        unsigned klds = lds_addr32(dstbuf);
        unsigned long long kg = (unsigned long long)(kh + (size_t)t0 * dh);
        unsigned vo = (unsigned)(lane * 128);
        unsigned dd = klds + vo + (unsigned)((lane >> 1) * 16);
        asm volatile(
            "global_load_async_to_lds_b128 %0, %1, %2 offset:0\n\t"
            "global_load_async_to_lds_b128 %0, %1, %2 offset:16\n\t"
            "global_load_async_to_lds_b128 %0, %1, %2 offset:32\n\t"
            "global_load_async_to_lds_b128 %0, %1, %2 offset:48\n\t"
            "global_load_async_to_lds_b128 %0, %1, %2 offset:64\n\t"
            "global_load_async_to_lds_b128 %0, %1, %2 offset:80\n\t"
            "global_load_async_to_lds_b128 %0, %1, %2 offset:96\n\t"
            "global_load_async_to_lds_b128 %0, %1, %2 offset:112"
            :: "v"(dd), "v"(vo), "s"(kg)
            : "memory");
    };

    // Phase 1: scores into LDS, double-buffered K pipeline (8 issues/tile;
    // async loads complete in order -> s_wait_asynccnt 8 drains current tile
    // while the next one is still in flight).
    {
        int buf = 0;
        issueK(w * 16, smKw);
        for (int ct = w; ct < Ss / 16; ct += 4) {
            const int t0 = ct * 16;
            const bool more = (ct + 4 < Ss / 16);
            if (more) issueK((ct + 4) * 16, smKw + (buf ^ 1) * 16 * QKSTRIDE);
            if (more) asm volatile("s_wait_asynccnt 0x8" ::: "memory");
            else      asm volatile("s_wait_asynccnt 0x0" ::: "memory");
            const float* smKc = smKw + buf * 16 * QKSTRIDE;
            const float* qrow = smQ + r * QKSTRIDE + 2 * hf;
            const float* krow = smKc + r * QKSTRIDE + 2 * hf;
            v8f acc = {};
#pragma unroll
            for (int k0 = 0; k0 < dh; k0 += 4) {
                v2f a = *(const v2f*)(qrow + k0);
                v2f b = *(const v2f*)(krow + k0);
                acc = WMMA_F32(a, b, acc);
            }
            for (int i = 0; i < 8; ++i)
                e[(i + 8 * hf) * ESTRIDE + t0 + r] = acc[i] * scale;
            buf ^= 1;
        }
    }
    __syncthreads();

    // Phase 2a: row mean / unbiased variance (8 threads per row)
    const int row = tid >> 3;
    const int sub = tid & 7;
    {
        float s1 = 0.f, s2 = 0.f;
        for (int j = sub; j < Ss; j += 8) {
            float v = e[row * ESTRIDE + j];
            s1 += v;
            s2 += v * v;
        }
        red[row * 8 + sub] = s1;
        red[128 + row * 8 + sub] = s2;
    }
    __syncthreads();
    if (tid < 16) {
        float s1 = 0.f, s2 = 0.f;
        for (int j = 0; j < 8; ++j) {
            s1 += red[tid * 8 + j];
            s2 += red[128 + tid * 8 + j];
        }
        float mu = s1 / (float)Ss;
        float var = (s2 - s1 * s1 / (float)Ss) / (float)(Ss - 1);  // ddof=1
        rowMu[tid] = mu;
        rowRs[tid] = rsqrtf(var + 1e-8f);
    }
    __syncthreads();

    // Phase 2b: layernorm + row max
    {
        const float g = gamma_a[h];
        const float bt = beta_a[h];
        const float mu = rowMu[row];
        const float rs = rowRs[row];
        float mx = -3.4e38f;
        for (int j = sub; j < Ss; j += 8) {
            float t = g * (e[row * ESTRIDE + j] - mu) * rs + bt;
            e[row * ESTRIDE + j] = t;
            mx = fmaxf(mx, t);
        }
        red[row * 8 + sub] = mx;
    }
    __syncthreads();
    if (tid < 16) {
        float mx = red[tid * 8];
        for (int j = 1; j < 8; ++j) mx = fmaxf(mx, red[tid * 8 + j]);
        rowMx[tid] = mx;
    }
    __syncthreads();

    // Phase 2c: exp + row sum
    {
        const float mx = rowMx[row];
        float s = 0.f;
        for (int j = sub; j < Ss; j += 8) {
            float t = expf(e[row * ESTRIDE + j] - mx);
            e[row * ESTRIDE + j] = t;
            s += t;
        }
        red[row * 8 + sub] = s;
    }
    __syncthreads();
    if (tid < 16) {
        float s = 0.f;
        for (int j = 0; j < 8; ++j) s += red[tid * 8 + j];
        rowSm[tid] = s;
    }
    __syncthreads();

    // Phase 3: heads = (exp @ V) / rowsum ; wave w -> head-dim cols [w*16,..)
    {
        const int n0 = w * 16;
        const float* erow = e + r * ESTRIDE + 2 * hf;
        v8f acc = {};
#pragma unroll 4
        for (int k0 = 0; k0 < Ss; k0 += 4) {
            v2f a = *(const v2f*)(erow + k0);
            v2f b;
            b.x = vh[(size_t)(k0 + 2 * hf) * dh + n0 + r];
            b.y = vh[(size_t)(k0 + 2 * hf + 1) * dh + n0 + r];
            acc = WMMA_F32(a, b, acc);
        }
        for (int i = 0; i < 8; ++i) {
            const int m = i + 8 * hf;
            const int sI = s0 + m;
            ab[((size_t)bI * Ss + sI) * DhT + h * dh + n0 + r] = acc[i] / rowSm[m];
        }
    }
}

// ---------------------------------------------------------------------------
// Kernel 3: x1 = a @ WO, out = x1 + x.   a:[8192,512], WO:[512,300].
// ---------------------------------------------------------------------------
__global__ void wo_kernel(const float* __restrict__ ab,
                          const float* __restrict__ WO,
                          const float* __restrict__ x,
                          float* __restrict__ out) {
    extern __shared__ float smA[];
    const int tid = threadIdx.x;
    const int lane = tid & 31, w = tid >> 5;
    const int r = lane & 15, hf = lane >> 4;
    const int m0 = blockIdx.x * 16;
    const int n0 = blockIdx.y * 64 + w * 16;

    if (tid < 32) {
        tdm_load_rows16(lds_addr32(smA), ab + (size_t)m0 * DhT, DhT, LDA_512);
        __builtin_amdgcn_s_wait_tensorcnt(0);
    }
    __syncthreads();

    if (n0 >= Dd) return;  // wave-uniform skip (tail tiles)
    const int c = n0 + r;
    const int cc = (c < Dd) ? c : (Dd - 1);
    v8f acc = wmma_ldsA_loop<DhT, LDA_512, Dd>(smA, WO + cc, r, hf);

    if (c < Dd) {
        for (int i = 0; i < 8; ++i) {
            int m = m0 + i + 8 * hf;
            out[(size_t)m * Dd + c] = acc[i] + x[(size_t)m * Dd + c];
        }
    }
}

// ---------------------------------------------------------------------------
// Kernel 4: h1 = relu(x2 @ W1 + b1).  x2:[8192,300], W1:[300,1200].
// ---------------------------------------------------------------------------
__global__ void ffn1_kernel(const float* __restrict__ x2,
                            const float* __restrict__ W1,
                            const float* __restrict__ b1,
                            float* __restrict__ h1) {
    extern __shared__ float smA[];
    const int tid = threadIdx.x;
    const int lane = tid & 31, w = tid >> 5;
    const int r = lane & 15, hf = lane >> 4;
    const int m0 = blockIdx.x * 16;
    const int n0 = blockIdx.y * 64 + w * 16;

    if (tid < 32) {
        tdm_load_rows16(lds_addr32(smA), x2 + (size_t)m0 * Dd, Dd, LDA_300);
        __builtin_amdgcn_s_wait_tensorcnt(0);
    }
    __syncthreads();

    if (n0 >= Df) return;  // wave-uniform skip
    const int c = n0 + r;  // always < Df for surviving waves
    v8f acc = wmma_ldsA_loop<Dd, LDA_300, Df>(smA, W1 + c, r, hf);

    const float bias = b1[c];
    for (int i = 0; i < 8; ++i) {
        int m = m0 + i + 8 * hf;
        h1[(size_t)m * Df + c] = fmaxf(acc[i] + bias, 0.f);
    }
}

// ---------------------------------------------------------------------------
// Kernel 5: out = h1 @ W2 + b2 + x2.  h1:[8192,1200], W2:[1200,300].
// ---------------------------------------------------------------------------
__global__ void ffn2_kernel(const float* __restrict__ h1,
                            const float* __restrict__ W2,
                            const float* __restrict__ b2,
                            const float* __restrict__ x2,
                            float* __restrict__ out) {
    extern __shared__ float smA[];
    const int tid = threadIdx.x;
    const int lane = tid & 31, w = tid >> 5;
    const int r = lane & 15, hf = lane >> 4;
    const int m0 = blockIdx.x * 16;
    const int n0 = blockIdx.y * 64 + w * 16;

    if (tid < 32) {
        tdm_load_rows16(lds_addr32(smA), h1 + (size_t)m0 * Df, Df, LDA_1200);
        __builtin_amdgcn_s_wait_tensorcnt(0);
    }
    __syncthreads();

    if (n0 >= Dd) return;  // wave-uniform skip
    const int c = n0 + r;
    const int cc = (c < Dd) ? c : (Dd - 1);
    v8f acc = wmma_ldsA_loop<Df, LDA_1200, Dd>(smA, W2 + cc, r, hf);

    if (c < Dd) {
        const float bias = b2[c];
        for (int i = 0; i < 8; ++i) {
            int m = m0 + i + 8 * hf;
            out[(size_t)m * Dd + c] = acc[i] + bias + x2[(size_t)m * Dd + c];
        }
    }
}

// ---------------------------------------------------------------------------
// LayerNorm over last dim (unbiased variance, scalar gamma/beta).
// ---------------------------------------------------------------------------
__global__ void ln_kernel(const float* __restrict__ in, float* __restrict__ out,
                          int cols, const float* __restrict__ gamma,
                          const float* __restrict__ beta) {
    __shared__ float r1[256];
    __shared__ float r2[256];
    __shared__ float mu_s, rs_s;
    const int row = blockIdx.x;
    const int tid = threadIdx.x;
    const float* p = in + (size_t)row * cols;
    float s1 = 0.f, s2 = 0.f;
    for (int j = tid; j < cols; j += 256) {
        float v = p[j];
        s1 += v;
        s2 += v * v;
    }
    r1[tid] = s1;
    r2[tid] = s2;
    __syncthreads();
    for (int st = 128; st > 0; st >>= 1) {
        if (tid < st) {
            r1[tid] += r1[tid + st];
            r2[tid] += r2[tid + st];
        }
        __syncthreads();
    }
    if (tid == 0) {
        float mu = r1[0] / (float)cols;
        float var = (r2[0] - r1[0] * r1[0] / (float)cols) / (float)(cols - 1);
        mu_s = mu;
        rs_s = rsqrtf(var + 1e-8f);
    }
    __syncthreads();
    const float g = gamma[0];
    const float bt = beta[0];
    float* q = out + (size_t)row * cols;
    for (int j = tid; j < cols; j += 256)
        q[j] = g * (p[j] - mu_s) * rs_s + bt;
}

// ---------------------------------------------------------------------------
extern "C" void kernel_launch(void* const* d_in, const int* in_sizes, int n_in,
                              void* d_out, int out_size, void* d_ws, size_t ws_size,
                              hipStream_t stream) {
    const float* x       = (const float*)d_in[0];
    const float* WQ      = (const float*)d_in[1];
    const float* WK      = (const float*)d_in[2];
    const float* WV      = (const float*)d_in[3];
    const float* WO      = (const float*)d_in[4];
    const float* W1      = (const float*)d_in[5];
    const float* b1      = (const float*)d_in[6];
    const float* W2      = (const float*)d_in[7];
    const float* b2      = (const float*)d_in[8];
    const float* gamma_a = (const float*)d_in[9];
    const float* beta_a  = (const float*)d_in[10];
    const float* gamma1  = (const float*)d_in[11];
    const float* beta1   = (const float*)d_in[12];
    const float* gamma2  = (const float*)d_in[13];
    const float* beta2   = (const float*)d_in[14];

    float* ws = (float*)d_ws;
    float* qb = ws;                               // [H,B,S,dh]
    float* kb = qb + (size_t)Hh * Bb * Ss * dh;
    float* vb = kb + (size_t)Hh * Bb * Ss * dh;
    float* ab = vb + (size_t)Hh * Bb * Ss * dh;   // [B,S,512]
    float* t1 = ab + (size_t)Mm * DhT;            // [8192,300] x1+x / ffn2 out
    float* x2 = t1 + (size_t)Mm * Dd;             // [8192,300]
    float* h1 = x2 + (size_t)Mm * Dd;             // [8192,1200]

    const dim3 blk128(128), blk256(256);

    // 1) QKV projections (TDM-staged A strip, head-split on write)
    qkv_kernel<<<dim3(Mm / 16, DhT / 64, 3), blk128,
                 (16 * LDA_300 + 16) * sizeof(float), stream>>>(
        x, WQ, WK, WV, qb, kb, vb);

    // 2) attention (async-staged Q + double-buffered K, LN->softmax->AV fused)
    const size_t smem_attn =
        (16 * ESTRIDE + 256 + 64 + 16 * QKSTRIDE + 4 * 2 * 16 * QKSTRIDE) *
        sizeof(float);
    attn_kernel<<<dim3(Ss / 16, Bb, Hh), blk128, smem_attn, stream>>>(
        qb, kb, vb, gamma_a, beta_a, ab);

    // 3) output projection + residual
    wo_kernel<<<dim3(Mm / 16, (Dd + 63) / 64), blk128,
                (16 * LDA_512 + 16) * sizeof(float), stream>>>(ab, WO, x, t1);

    // 4) layernorm 1
    ln_kernel<<<dim3(Mm), blk256, 0, stream>>>(t1, x2, Dd, gamma1, beta1);

    // 5) FFN
    ffn1_kernel<<<dim3(Mm / 16, (Df + 63) / 64), blk128,
                  (16 * LDA_300 + 16) * sizeof(float), stream>>>(x2, W1, b1, h1);
    ffn2_kernel<<<dim3(Mm / 16, (Dd + 63) / 64), blk128,
                  (16 * LDA_1200 + 16) * sizeof(float), stream>>>(h1, W2, b2, x2, t1);

    // 6) layernorm 2 -> output
    ln_kernel<<<dim3(Mm), blk256, 0, stream>>>(t1, (float*)d_out, Dd, gamma2, beta2);
}